// SparsifyAbs2d_39109972198313
// MI455X (gfx1250) — compile-verified
//
#include <hip/hip_runtime.h>

typedef unsigned int u32;
typedef u32 u32x4 __attribute__((ext_vector_type(4)));

#define TPB   256          // 8 waves of 32; also == number of radix bins
#define ROW   16384        // H*W elements per (b,c) row
#define KEEP  8192         // k = 0.5 * H * W
#define VEC   16           // uint4 chunks per thread: ROW / (TPB*4)
#define NH0   32           // pass-0 private hists (hot exponent digit)
#define NH1   4            // pass-1..3 private hists (uniform digits)
#define HPAD  257          // 256 bins + 1 pad dword -> bank = (group + digit) % 64
#define NWAVE (TPB / 32)

__global__ __launch_bounds__(TPB) void SparsifyAbs2d_kernel(const u32* __restrict__ x,
                                                            u32* __restrict__ out) {
    __shared__ u32 vals[ROW];                       // 64 KB: one row in LDS
    __shared__ __align__(16) u32 hist[NH0][HPAD];   // 32.9 KB bank-skewed hists
    __shared__ u32 wtot[NWAVE];                     // per-wave bin-count totals
    __shared__ u32 sel_digit;
    __shared__ u32 sel_remaining;

    const u32 tid  = threadIdx.x;                   // also: my radix bin
    const u32 lane = tid & 31u;
    const u32 wave = tid >> 5;                      // wave32
    const u32 row  = blockIdx.x;

    const u32* gin  = x   + (size_t)row * ROW;
    u32*       gout = out + (size_t)row * ROW;

    // -------- Stage the row into LDS: CDNA5 async global->LDS, GVS mode -----
    // INST_OFFSET applies to both the LDS and the global address.
    const u32 lds_addr = (u32)(size_t)&vals[0] + tid * 16u;
    const u32 vofs     = tid * 16u;
    const unsigned long long sbase = (unsigned long long)gin;
    #pragma unroll
    for (int it = 0; it < VEC; ++it) {
        asm volatile("global_load_async_to_lds_b128 %0, %1, %2 offset:%3"
                     :
                     : "v"(lds_addr), "v"(vofs), "s"(sbase), "i"(it * TPB * 16)
                     : "memory");
    }
    asm volatile("s_wait_asynccnt 0" ::: "memory");  // this wave's copies done
    __syncthreads();                                 // all waves' copies visible

    // -------- 4-pass MSB-first radix select of the k-th largest |x| ---------
    u32 prefix    = 0u;     // known high bits of threshold (abs bit pattern)
    u32 mask      = 0u;     // which bits of prefix are valid
    u32 remaining = KEEP;   // rank within the masked candidate set

    #pragma unroll
    for (int pass = 0; pass < 4; ++pass) {
        const u32 shift = 24u - (u32)pass * 8u;
        const u32 nh    = (pass == 0) ? NH0 : NH1;
        const u32 grp   = (pass == 0) ? (tid >> 3) : (tid >> 6);

        // clear private histograms with b128 stores (nh*257 % 4 == 0)
        {
            const u32 nvec = (nh * HPAD) >> 2;
            u32x4* hv = (u32x4*)&hist[0][0];
            const u32x4 z = {0u, 0u, 0u, 0u};
            for (u32 i = tid; i < nvec; i += TPB) hv[i] = z;
        }
        __syncthreads();

        // histogram current byte over candidates matching the known prefix
        #pragma unroll
        for (int it = 0; it < VEC; ++it) {
            const u32 base = (u32)(it * TPB + tid) * 4u;
            u32x4 v = *(const u32x4*)&vals[base];             // ds_load_b128
            u32 a;
            a = v.x & 0x7fffffffu;
            if ((a & mask) == prefix) atomicAdd(&hist[grp][(a >> shift) & 0xffu], 1u);
            a = v.y & 0x7fffffffu;
            if ((a & mask) == prefix) atomicAdd(&hist[grp][(a >> shift) & 0xffu], 1u);
            a = v.z & 0x7fffffffu;
            if ((a & mask) == prefix) atomicAdd(&hist[grp][(a >> shift) & 0xffu], 1u);
            a = v.w & 0x7fffffffu;
            if ((a & mask) == prefix) atomicAdd(&hist[grp][(a >> shift) & 0xffu], 1u);
        }
        __syncthreads();

        // my bin's total over the private histograms (conflict-free reads)
        u32 myc = 0u;
        for (u32 g = 0; g < nh; ++g) myc += hist[g][tid];

        // wave32 in-lane suffix scan: s = sum of myc over lanes >= lane
        u32 s = myc;
        #pragma unroll
        for (u32 d = 1u; d < 32u; d <<= 1) {
            const u32 t = __shfl_down(s, d);
            if (lane + d < 32u) s += t;
        }
        if (lane == 0u) wtot[wave] = s;   // total of this wave's 32 bins
        __syncthreads();

        // counts from higher-numbered waves (higher bins)
        u32 hi = 0u;
        #pragma unroll
        for (u32 w = 0u; w < NWAVE; ++w)
            if (w > wave) hi += wtot[w];

        // ssum[d] = # candidates with digit >= d; pick unique d with
        // ssum[d] >= remaining > ssum[d+1]
        const u32 snext  = __shfl_down(s, 1);            // before divergence
        const u32 ssum_d = s + hi;
        const u32 nxt    = (lane == 31u) ? hi : (snext + hi);
        if (ssum_d >= remaining && nxt < remaining) {
            sel_digit     = tid;
            sel_remaining = remaining - nxt;             // rank within bin
        }
        __syncthreads();   // broadcast selection to all threads

        prefix   |= sel_digit << shift;
        mask     |= 0xffu << shift;
        remaining = sel_remaining;
        // next pass's hist-clear is safe without another barrier: everyone's
        // hist reads (myc) happened before the wtot barrier above.
    }

    // prefix now holds the exact abs-bit pattern of the k-th largest |x|
    // (uint compare on abs bits == float compare for non-negative floats).
    const u32 thresh = prefix;

    // -------- Output pass: keep |x| >= thresh (ties kept), else signed zero --
    // NT stores: 256 MB written once, never reread, exceeds the 192 MB L2.
    #pragma unroll
    for (int it = 0; it < VEC; ++it) {
        const u32 base = (u32)(it * TPB + tid) * 4u;
        u32x4 v = *(const u32x4*)&vals[base];
        u32x4 o;
        o.x = ((v.x & 0x7fffffffu) >= thresh) ? v.x : (v.x & 0x80000000u);
        o.y = ((v.y & 0x7fffffffu) >= thresh) ? v.y : (v.y & 0x80000000u);
        o.z = ((v.z & 0x7fffffffu) >= thresh) ? v.z : (v.z & 0x80000000u);
        o.w = ((v.w & 0x7fffffffu) >= thresh) ? v.w : (v.w & 0x80000000u);
        __builtin_nontemporal_store(o, (u32x4*)&gout[base]);  // b128 NT store
    }
}

extern "C" void kernel_launch(void* const* d_in, const int* in_sizes, int n_in,
                              void* d_out, int out_size, void* d_ws, size_t ws_size,
                              hipStream_t stream) {
    const u32* x = (const u32*)d_in[0];
    u32* out = (u32*)d_out;
    const int rows = in_sizes[0] / ROW;   // B*C = 4096 for the reference shape
    SparsifyAbs2d_kernel<<<rows, TPB, 0, stream>>>(x, out);
}